// Detr3DCrossAtten_78864189489567
// MI455X (gfx1250) — compile-verified
//
#include <hip/hip_runtime.h>
#include <hip/hip_bf16.h>
#include <math.h>

// ---------------------------------------------------------------------------
// Detr3D cross-attention for MI455X (gfx1250, wave32).
//  - Dense layers via V_WMMA_F32_16X16X4_F32; each wave owns 4 adjacent 16x16
//    C tiles (A-fragment reused 4x, 8 WMMAs per unrolled K iteration, no
//    branches in the inner loop).
//  - Feature pyramids transposed NCHW -> NHWC once so each bilinear tap is a
//    contiguous 1KB read (189MB pyramid ~fits the 192MB L2); falls back to
//    strided gather if the workspace is too small.
// ---------------------------------------------------------------------------

typedef float v2f __attribute__((ext_vector_type(2)));
typedef float v8f __attribute__((ext_vector_type(8)));

#define QN   1024
#define CN   256
#define NCAM 6
#define NLVL 4

// --------------------------- WMMA f32 GEMM ---------------------------------
// C[M,N] = (A (+A2)) [M,K] @ B [K,N] + bias.  Row-major. N % (64*TPW) == 0,
// K % 4 == 0, M % 16 == 0 (all true for this model).
// Block = 128 threads = 4 waves; wave w owns cols [w*16*TPW, w*16*TPW+16*TPW).
// A-frag (16x4 f32): lanes 0-15 hold K=k0,k0+1 ; lanes 16-31 hold K=k0+2,k0+3.
// B-frag (4x16 f32): mirrored split across lane halves.
// D layout: VGPR r -> M = r + 8*(lane>=16), N = lane&15.
template <bool HAS_A2, int TPW>
__global__ __launch_bounds__(128)
void k_gemm_wmma(const float* __restrict__ A, const float* __restrict__ A2,
                 const float* __restrict__ Bm, const float* __restrict__ bias,
                 float* __restrict__ Cm, int K, int N) {
  const int lane = threadIdx.x & 31;
  const int wave = threadIdx.x >> 5;
  const int half = lane >> 4;
  const int lr   = lane & 15;
  const int row0 = blockIdx.x * 16;
  const int colw = blockIdx.y * (64 * TPW) + wave * (16 * TPW) + lr;
  const int row  = row0 + lr;

  const float* __restrict__ Arow  = A  + (size_t)row * K;
  const float* __restrict__ A2row = A2 + (size_t)row * K;

  v8f acc[TPW];
  {
    v8f z = {};
#pragma unroll
    for (int t = 0; t < TPW; ++t) acc[t] = z;
  }

#pragma unroll 2
  for (int k0 = 0; k0 < K; k0 += 4) {
    const int ka = k0 + 2 * half;          // even -> 8B-aligned b64 load
    v2f a = *(const v2f*)(Arow + ka);
    if (HAS_A2) {
      v2f a2 = *(const v2f*)(A2row + ka);
      a.x += a2.x; a.y += a2.y;
    }
    const float* __restrict__ Brow0 = Bm + (size_t)ka * N + colw;
    const float* __restrict__ Brow1 = Brow0 + N;
#pragma unroll
    for (int t = 0; t < TPW; ++t) {
      v2f b;
      b.x = Brow0[t * 16];
      b.y = Brow1[t * 16];
      acc[t] = __builtin_amdgcn_wmma_f32_16x16x4_f32(
          /*neg_a=*/false, a, /*neg_b=*/false, b,
          /*c_mod=*/(short)0, acc[t], /*reuse_a=*/false, /*reuse_b=*/false);
    }
  }

#pragma unroll
  for (int t = 0; t < TPW; ++t) {
    const int col = colw + t * 16;
    const float bv = bias[col];
#pragma unroll
    for (int r = 0; r < 8; ++r)
      Cm[(size_t)(row0 + r + 8 * half) * N + col] = acc[t][r] + bv;
  }
}

// ---------------------- camera projection / grid ---------------------------
__global__ void k_prepare(const float* __restrict__ rp,
                          const float* __restrict__ l2i,
                          float* __restrict__ gx, float* __restrict__ gy,
                          float* __restrict__ mv) {
  int t = blockIdx.x * blockDim.x + threadIdx.x;
  if (t >= NCAM * QN) return;
  int n = t / QN, q = t % QN;
  float p0 = rp[q * 3 + 0], p1 = rp[q * 3 + 1], p2 = rp[q * 3 + 2];
  const float* M = l2i + n * 16;
  float r0 = M[0] * p0 + M[1] * p1 + M[2]  * p2 + M[3];
  float r1 = M[4] * p0 + M[5] * p1 + M[6]  * p2 + M[7];
  float r2 = M[8] * p0 + M[9] * p1 + M[10] * p2 + M[11];
  bool front = r2 > 1e-5f;
  float zz = fmaxf(r2, 1e-5f);
  float px = r0 / zz, py = r1 / zz;
  float ggx = (px * (1.0f / 1600.0f) - 0.5f) * 2.0f;
  float ggy = (py * (1.0f / 928.0f)  - 0.5f) * 2.0f;
  bool inb = (ggx > -1.0f) && (ggx < 1.0f) && (ggy > -1.0f) && (ggy < 1.0f);
  gx[t] = ggx; gy[t] = ggy;
  mv[t] = (front && inb) ? 1.0f : 0.0f;
}

// ------------------- attention weights: sigmoid(qe@Wa+b)*mask --------------
__global__ void k_attw(const float* __restrict__ qe, const float* __restrict__ Wa,
                       const float* __restrict__ ba, const float* __restrict__ mv,
                       float* __restrict__ attw) {
  int t = blockIdx.x * blockDim.x + threadIdx.x;
  if (t >= QN * 24) return;
  int q = t / 24, j = t % 24;
  float acc = ba[j];
  const float* row = qe + (size_t)q * CN;
  for (int k = 0; k < CN; ++k) acc += row[k] * Wa[(size_t)k * 24 + j];
  float sgm = 1.0f / (1.0f + __expf(-acc));
  int n = j >> 2;  // j = n*4 + level
  attw[t] = sgm * mv[n * QN + q];
}

// ----------------------- NCHW -> NHWC transpose ----------------------------
__global__ __launch_bounds__(1024)
void k_transpose(const float* __restrict__ src, float* __restrict__ dst, int HW) {
  __shared__ float tile[32][33];
  int n  = blockIdx.z;
  int s  = blockIdx.x * 32 + threadIdx.x;
  int ci = blockIdx.y * 32 + threadIdx.y;
  if (s < HW) tile[threadIdx.y][threadIdx.x] = src[((size_t)(n * CN + ci)) * HW + s];
  __syncthreads();
  int s2 = blockIdx.x * 32 + threadIdx.y;
  int c2 = blockIdx.y * 32 + threadIdx.x;
  if (s2 < HW) dst[((size_t)(n * HW + s2)) * CN + c2] = tile[threadIdx.x][threadIdx.y];
}

// ------------------- fused bilinear sample + weighted sum ------------------
struct FeatPtrs {
  const float* f[NLVL];   // original (N,C,H,W)
  const float* t[NLVL];   // transposed (N,H,W,C)
};

__global__ __launch_bounds__(256)
void k_sample(FeatPtrs fp, const float* __restrict__ gxv,
              const float* __restrict__ gyv, const float* __restrict__ mv,
              const float* __restrict__ attw, float* __restrict__ mix,
              int transposed) {
  const int q = blockIdx.x;
  const int c = threadIdx.x;
  const int HS[NLVL] = {116, 58, 29, 15};
  const int WS[NLVL] = {200, 100, 50, 25};
  float acc = 0.0f;
  for (int n = 0; n < NCAM; ++n) {
    if (mv[n * QN + q] == 0.0f) continue;     // uniform across block
    const float gx = gxv[n * QN + q];
    const float gy = gyv[n * QN + q];
#pragma unroll
    for (int l = 0; l < NLVL; ++l) {
      const int H = HS[l], W = WS[l], HW = H * W;
      float x = ((gx + 1.0f) * W - 1.0f) * 0.5f;
      float y = ((gy + 1.0f) * H - 1.0f) * 0.5f;
      float fx0 = floorf(x), fy0 = floorf(y);
      int x0 = (int)fx0, y0 = (int)fy0;
      float wx1 = x - fx0, wy1 = y - fy0;
      float wx0 = 1.0f - wx1, wy0 = 1.0f - wy1;
      const float wl = attw[q * 24 + n * 4 + l];
      const float tw[4] = {wy0 * wx0, wy0 * wx1, wy1 * wx0, wy1 * wx1};
      const int ix[4] = {x0, x0 + 1, x0,     x0 + 1};
      const int iy[4] = {y0, y0,     y0 + 1, y0 + 1};
#pragma unroll
      for (int tp = 0; tp < 4; ++tp) {
        if (ix[tp] < 0 || ix[tp] >= W || iy[tp] < 0 || iy[tp] >= H) continue;
        const int idx = iy[tp] * W + ix[tp];
        float v = transposed
                    ? fp.t[l][((size_t)(n * HW + idx)) * CN + c]     // 1KB coalesced
                    : fp.f[l][((size_t)(n * CN + c)) * HW + idx];    // strided fallback
        acc += wl * tw[tp] * v;
      }
    }
  }
  mix[(size_t)q * CN + c] = acc;
}

// ------------------------------ layernorm ----------------------------------
__device__ __forceinline__ float block_reduce_sum(float v, float* sbuf) {
  int t = threadIdx.x;
  sbuf[t] = v; __syncthreads();
  for (int s = blockDim.x >> 1; s > 0; s >>= 1) {
    if (t < s) sbuf[t] += sbuf[t + s];
    __syncthreads();
  }
  float r = sbuf[0]; __syncthreads();
  return r;
}

// pos = relu(LN(rp @ pe_w1 + pe_b1))  (K=3 -> plain FMA)
__global__ __launch_bounds__(256)
void k_pos1(const float* __restrict__ rp, const float* __restrict__ w1,
            const float* __restrict__ b1, const float* __restrict__ g1,
            const float* __restrict__ be1, float* __restrict__ out) {
  __shared__ float sbuf[CN];
  int q = blockIdx.x, c = threadIdx.x;
  float x = rp[q * 3 + 0], y = rp[q * 3 + 1], z = rp[q * 3 + 2];
  float v = x * w1[c] + y * w1[CN + c] + z * w1[2 * CN + c] + b1[c];
  float s  = block_reduce_sum(v, sbuf);
  float s2 = block_reduce_sum(v * v, sbuf);
  float mean = s * (1.0f / CN);
  float var  = s2 * (1.0f / CN) - mean * mean;
  float r = (v - mean) * rsqrtf(var + 1e-5f) * g1[c] + be1[c];
  out[(size_t)q * CN + c] = fmaxf(r, 0.0f);
}

__global__ void k_ln(const float* __restrict__ in, const float* __restrict__ g,
                     const float* __restrict__ b, float* __restrict__ out,
                     int C, int relu) {
  extern __shared__ float sbuf[];
  int q = blockIdx.x, c = threadIdx.x;
  float v  = in[(size_t)q * C + c];
  float s  = block_reduce_sum(v, sbuf);
  float s2 = block_reduce_sum(v * v, sbuf);
  float inv = 1.0f / (float)C;
  float mean = s * inv;
  float var  = s2 * inv - mean * mean;
  float r = (v - mean) * rsqrtf(var + 1e-5f) * g[c] + b[c];
  if (relu) r = fmaxf(r, 0.0f);
  out[(size_t)q * C + c] = r;
}

__global__ void k_add3(const float* __restrict__ a, const float* __restrict__ b,
                       const float* __restrict__ c, float* __restrict__ o, int n) {
  int t = blockIdx.x * blockDim.x + threadIdx.x;
  if (t < n) o[t] = a[t] + b[t] + c[t];
}

// ------------------------------- host --------------------------------------
extern "C" void kernel_launch(void* const* d_in, const int* in_sizes, int n_in,
                              void* d_out, int out_size, void* d_ws, size_t ws_size,
                              hipStream_t stream) {
  (void)in_sizes; (void)n_in; (void)out_size;
  const float* query     = (const float*)d_in[0];
  const float* query_pos = (const float*)d_in[1];
  const float* ref_pts   = (const float*)d_in[2];
  const float* lidar2img = (const float*)d_in[3];
  const float* feats[NLVL] = {(const float*)d_in[4], (const float*)d_in[5],
                              (const float*)d_in[6], (const float*)d_in[7]};
  const float* W_qe   = (const float*)d_in[8];
  const float* b_qe   = (const float*)d_in[9];
  const float* W_attn = (const float*)d_in[10];
  const float* b_attn = (const float*)d_in[11];
  const float* W_out  = (const float*)d_in[12];
  const float* b_out  = (const float*)d_in[13];
  const float* pe_w1  = (const float*)d_in[14];
  const float* pe_b1  = (const float*)d_in[15];
  const float* pe_g1  = (const float*)d_in[16];
  const float* pe_be1 = (const float*)d_in[17];
  const float* pe_w2  = (const float*)d_in[18];
  const float* pe_b2  = (const float*)d_in[19];
  const float* pe_g2  = (const float*)d_in[20];
  const float* pe_be2 = (const float*)d_in[21];
  const float* W_fin  = (const float*)d_in[22];
  const float* b_fin  = (const float*)d_in[23];
  const float* g_norm = (const float*)d_in[24];
  const float* b_norm = (const float*)d_in[25];
  float* outp = (float*)d_out;

  // ---- workspace carve-up (floats) ----
  float* w = (float*)d_ws;
  size_t o = 0;
  auto alloc = [&](size_t n) { float* p = w + o; o += n; return p; };
  float* qe    = alloc((size_t)QN * CN);
  float* attw  = alloc((size_t)QN * 24);
  float* gxb   = alloc((size_t)NCAM * QN);
  float* gyb   = alloc((size_t)NCAM * QN);
  float* mvb   = alloc((size_t)NCAM * QN);
  float* mix   = alloc((size_t)QN * CN);
  float* outb  = alloc((size_t)QN * CN);
  float* pos1  = alloc((size_t)QN * CN);
  float* pos2t = alloc((size_t)QN * CN);
  float* pos2  = alloc((size_t)QN * CN);
  float* sum3  = alloc((size_t)QN * CN);
  float* fin   = alloc((size_t)QN * 64);
  const size_t hw[NLVL] = {116 * 200, 58 * 100, 29 * 50, 15 * 25};
  float* ftr[NLVL];
  for (int l = 0; l < NLVL; ++l) ftr[l] = alloc((size_t)NCAM * hw[l] * CN);
  const int transposed = (o * sizeof(float) <= ws_size) ? 1 : 0;

  // 1) projection / grid / mask
  k_prepare<<<(NCAM * QN + 255) / 256, 256, 0, stream>>>(ref_pts, lidar2img, gxb, gyb, mvb);

  // 2) qe = (query + query_pos) @ W_qe + b_qe   [1024x64 @ 64x256]   (WMMA)
  k_gemm_wmma<true, 4><<<dim3(QN / 16, 1), 128, 0, stream>>>(
      query, query_pos, W_qe, b_qe, qe, 64, CN);

  // 3) attention weights: sigmoid(qe @ W_attn + b) * mask
  k_attw<<<(QN * 24 + 255) / 256, 256, 0, stream>>>(qe, W_attn, b_attn, mvb, attw);

  // 4) NCHW -> NHWC pyramid transpose (if workspace allows)
  if (transposed) {
    for (int l = 0; l < NLVL; ++l) {
      dim3 g((unsigned)((hw[l] + 31) / 32), CN / 32, NCAM);
      k_transpose<<<g, dim3(32, 32), 0, stream>>>(feats[l], ftr[l], (int)hw[l]);
    }
  }

  // 5) fused bilinear sampling + (sigmoid*mask)-weighted reduction over (N,L)
  FeatPtrs fp;
  for (int l = 0; l < NLVL; ++l) { fp.f[l] = feats[l]; fp.t[l] = ftr[l]; }
  k_sample<<<QN, CN, 0, stream>>>(fp, gxb, gyb, mvb, attw, mix, transposed);

  // 6) out = mix @ W_out + b_out   [1024x256 @ 256x256]  (WMMA)
  k_gemm_wmma<false, 4><<<dim3(QN / 16, 1), 128, 0, stream>>>(
      mix, nullptr, W_out, b_out, outb, CN, CN);

  // 7) positional MLP
  k_pos1<<<QN, CN, 0, stream>>>(ref_pts, pe_w1, pe_b1, pe_g1, pe_be1, pos1);
  k_gemm_wmma<false, 4><<<dim3(QN / 16, 1), 128, 0, stream>>>(
      pos1, nullptr, pe_w2, pe_b2, pos2t, CN, CN);
  k_ln<<<QN, CN, CN * sizeof(float), stream>>>(pos2t, pe_g2, pe_be2, pos2, CN, 1);

  // 8) residual sum, final projection + layernorm
  k_add3<<<(QN * CN + 255) / 256, 256, 0, stream>>>(outb, qe, pos2, sum3, QN * CN);
  k_gemm_wmma<false, 1><<<dim3(QN / 16, 1), 128, 0, stream>>>(
      sum3, nullptr, W_fin, b_fin, fin, CN, 64);
  k_ln<<<QN, 64, 64 * sizeof(float), stream>>>(fin, g_norm, b_norm, outp, 64, 0);
}